// MoeLayer_76905684402186
// MI455X (gfx1250) — compile-verified
//
#include <hip/hip_runtime.h>

// ---------------------------------------------------------------------------
// MoE SwiGLU layer for MI455X (gfx1250, wave32, WMMA + async-LDS copies).
//
// Roofline: weights are 1.1 GB fp32 -> ~47us at 23.3 TB/s; dense compute over
// all 8 experts is ~2.3 TFLOP which fp32 WMMA (16x16x4) covers in single-digit
// us. => memory bound; fp32 everywhere (exact numerics), all experts computed
// densely (gate==0 kills unrouted pairs), weights streamed once.
// Staging pipeline uses GLOBAL_LOAD_ASYNC_TO_LDS_B128 (ASYNCcnt) with double
// buffering, 2-stage-unrolled steady state (no branches, static buffer ids).
// Per-expert h[T,H] (92 MB) is staged in workspace; per-expert working sets
// (w1+w3 = 92 MB, then h+w2 = 138 MB) fit the 192 MB L2.
// ---------------------------------------------------------------------------

#define T_TOK 4096
#define D_DIM 2048
#define H_DIM 5632
#define N_EXP 8

typedef __attribute__((ext_vector_type(2))) float v2f;
typedef __attribute__((ext_vector_type(8))) float v8f;

// Generic pointers into LDS carry the LDS byte offset in the low 32 bits
// (ISA 10.2 aperture rules), so truncation gives the DS address.
__device__ __forceinline__ unsigned lds_addr(const void* p) {
    return (unsigned)(uintptr_t)p;
}

// Async memory -> LDS copy of one 16B chunk per lane (tracked by ASYNCcnt).
__device__ __forceinline__ void async_copy16(unsigned lds, const void* g) {
    asm volatile("global_load_async_to_lds_b128 %0, %1, off"
                 :: "v"(lds), "v"(g) : "memory");
}
// Wait until at most N of this wave's async copies remain in flight.
// Async loads complete in order, so N == (copies issued for the next stage)
// guarantees the current stage has landed.
#define WAIT_ASYNC(n) asm volatile("s_wait_asynccnt " #n ::: "memory")

// ---------------------------------------------------------------------------
// Kernel 1: gating. One wave (32 lanes) per token: 8 logits via lane-strided
// dot products + shfl_xor reduction, then top-2 + softmax -> dense gate [T,E].
// ---------------------------------------------------------------------------
__global__ __launch_bounds__(256) void moe_gate_kernel(
    const float* __restrict__ x,   // [T, D]
    const float* __restrict__ Wg,  // [D, E]
    float* __restrict__ gate)      // [T, E] dense combine weights
{
    const int wave = threadIdx.x >> 5;
    const int lane = threadIdx.x & 31;
    const int t = blockIdx.x * 8 + wave;

    float acc[N_EXP];
#pragma unroll
    for (int e = 0; e < N_EXP; ++e) acc[e] = 0.0f;

    const float* xr = x + (size_t)t * D_DIM;
    for (int d = lane; d < D_DIM; d += 32) {
        const float xv = xr[d];
        const float* wr = Wg + d * N_EXP;
#pragma unroll
        for (int e = 0; e < N_EXP; ++e) acc[e] += xv * wr[e];
    }
#pragma unroll
    for (int e = 0; e < N_EXP; ++e) {
#pragma unroll
        for (int off = 16; off > 0; off >>= 1)
            acc[e] += __shfl_xor(acc[e], off, 32);
    }
    if (lane == 0) {
        int i0 = 0; float m0 = acc[0];
#pragma unroll
        for (int e = 1; e < N_EXP; ++e)
            if (acc[e] > m0) { m0 = acc[e]; i0 = e; }
        int i1 = (i0 == 0) ? 1 : 0; float m1 = acc[i1];
#pragma unroll
        for (int e = 0; e < N_EXP; ++e)
            if (e != i0 && acc[e] > m1) { m1 = acc[e]; i1 = e; }
        // softmax over the two kept logits (m0 >= m1, stable)
        const float e1 = __expf(m1 - m0);
        const float inv = 1.0f / (1.0f + e1);
        float* gr = gate + t * N_EXP;
#pragma unroll
        for (int e = 0; e < N_EXP; ++e) gr[e] = 0.0f;
        gr[i0] = inv;
        gr[i1] = e1 * inv;
    }
}

// ---------------------------------------------------------------------------
// Kernel 2: out = next_r  (residual init, vectorized)
// ---------------------------------------------------------------------------
__global__ __launch_bounds__(256) void moe_init_out(
    const float4* __restrict__ next_r, float4* __restrict__ out, int n4)
{
    const int i = blockIdx.x * 256 + threadIdx.x;
    if (i < n4) out[i] = next_r[i];
}

// ---------------------------------------------------------------------------
// WMMA f32 16x16x4 lane layouts (ISA 7.12.2, wave32):
//   A 16x4 : lane = half*16 + r (r = row M).  VGPR0 = A[r, k0+2*half],
//            VGPR1 = A[r, k0+2*half+1]
//   B 4x16 : lane c = lane&15 (col N). VGPR0 = B[k0+2*half, c],
//            VGPR1 = B[k0+2*half+1, c]   (row-striped like C/D)
//   C 16x16: VGPR i = C[i + 8*half, lane&15]
// ---------------------------------------------------------------------------
__device__ __forceinline__ v8f wmma_f32_4(v2f a, v2f b, v8f c) {
    return __builtin_amdgcn_wmma_f32_16x16x4_f32(false, a, false, b,
                                                 (short)0, c, false, false);
}

// ---------------------------------------------------------------------------
// Kernel 3 (per expert): h = silu(x @ w1) * (x @ w3).
// Block = 256 threads = 8 waves, output tile 64(M) x 64(N); waves 4(M)x2(N),
// each wave owns 16 rows x 32 cols = 2 C-tiles per weight matrix; w1/w3 share
// the A operand. K staged in steps of 16 through double-buffered LDS filled
// with async copies; steady-state loop unrolled over both buffers (branch-free
// body, compile-time buffer ids). LDS pads keep rows 16B-aligned (async
// copies are contiguous 16B chunks) and bank-conflict free.
// ---------------------------------------------------------------------------
__global__ __launch_bounds__(256) void moe_h_kernel(
    const float* __restrict__ x,    // [T, D]
    const float* __restrict__ w1,   // [D, H] (this expert)
    const float* __restrict__ w3,   // [D, H]
    float* __restrict__ h)          // [T, H]
{
    __shared__ float As[2][64][20];    // x tile   [M=64][K=16] (+4 pad)
    __shared__ float B1s[2][16][72];   // w1 tile  [K=16][N=64] (+8 pad)
    __shared__ float B3s[2][16][72];   // w3 tile

    const int n0 = blockIdx.x * 64;          // H offset
    const int m0 = blockIdx.y * 64;          // T offset
    const int tid = threadIdx.x;
    const int wave = tid >> 5, lane = tid & 31;
    const int wm = wave & 3;                 // wave row (16 rows each)
    const int wn = wave >> 2;                // wave col (32 cols each)
    const int r = lane & 15, half = lane >> 4;

    v8f acc1[2], acc3[2];
#pragma unroll
    for (int ns = 0; ns < 2; ++ns)
#pragma unroll
        for (int i = 0; i < 8; ++i) { acc1[ns][i] = 0.0f; acc3[ns][i] = 0.0f; }

    // one 16B async chunk per thread per tile per stage
    const int a_row = tid >> 2, a_col = (tid & 3) * 4;     // 64x16
    const int b_row = tid >> 4, b_col = (tid & 15) * 4;    // 16x64
    const float* gA  = x  + (size_t)(m0 + a_row) * D_DIM + a_col;   // +k0
    const float* gB1 = w1 + (size_t)b_row * H_DIM + n0 + b_col;     // +k0*H
    const float* gB3 = w3 + (size_t)b_row * H_DIM + n0 + b_col;
    const unsigned lA[2]  = { lds_addr(&As[0][a_row][a_col]),
                              lds_addr(&As[1][a_row][a_col]) };
    const unsigned lB1[2] = { lds_addr(&B1s[0][b_row][b_col]),
                              lds_addr(&B1s[1][b_row][b_col]) };
    const unsigned lB3[2] = { lds_addr(&B3s[0][b_row][b_col]),
                              lds_addr(&B3s[1][b_row][b_col]) };

    auto issue = [&](int buf, int k0) {
        async_copy16(lA[buf],  gA + k0);
        async_copy16(lB1[buf], gB1 + (size_t)k0 * H_DIM);
        async_copy16(lB3[buf], gB3 + (size_t)k0 * H_DIM);
    };
    auto compute = [&](int buf) {
#pragma unroll
        for (int kk = 0; kk < 16; kk += 4) {
            v2f a;
            a.x = As[buf][wm * 16 + r][kk + 2 * half];
            a.y = As[buf][wm * 16 + r][kk + 2 * half + 1];
#pragma unroll
            for (int ns = 0; ns < 2; ++ns) {
                const int c = wn * 32 + ns * 16 + r;
                v2f b1, b3;
                b1.x = B1s[buf][kk + 2 * half][c];
                b1.y = B1s[buf][kk + 2 * half + 1][c];
                b3.x = B3s[buf][kk + 2 * half][c];
                b3.y = B3s[buf][kk + 2 * half + 1][c];
                acc1[ns] = wmma_f32_4(a, b1, acc1[ns]);
                acc3[ns] = wmma_f32_4(a, b3, acc3[ns]);
            }
        }
    };

    // software pipeline: prologue
    issue(0, 0);
    // steady state, 2 stages per iteration (D/16 = 128 stages, even)
    int k0 = 0;
    for (; k0 + 32 < D_DIM; k0 += 32) {
        issue(1, k0 + 16); WAIT_ASYNC(3);
        __syncthreads(); compute(0); __syncthreads();
        issue(0, k0 + 32); WAIT_ASYNC(3);
        __syncthreads(); compute(1); __syncthreads();
    }
    // epilogue: k0 == D-32 here
    issue(1, k0 + 16); WAIT_ASYNC(3);
    __syncthreads(); compute(0); __syncthreads();
    WAIT_ASYNC(0);
    __syncthreads(); compute(1);

    // epilogue: silu(a1) * a3 -> h
#pragma unroll
    for (int i = 0; i < 8; ++i) {
        const int row = m0 + wm * 16 + i + 8 * half;
#pragma unroll
        for (int ns = 0; ns < 2; ++ns) {
            const int col = n0 + wn * 32 + ns * 16 + r;
            const float a = acc1[ns][i];
            const float hv = (a / (1.0f + __expf(-a))) * acc3[ns][i];
            h[(size_t)row * H_DIM + col] = hv;
        }
    }
}

// ---------------------------------------------------------------------------
// Kernel 4 (per expert): out[t, :] += gate[t, e] * (h @ w2).
// Same tiling; K = H = 5632 (352 stages, even). Non-atomic RMW on out is
// safe: per launch each element has exactly one writer and expert launches
// are stream-ordered.
// ---------------------------------------------------------------------------
__global__ __launch_bounds__(256) void moe_out_kernel(
    const float* __restrict__ h,     // [T, H]
    const float* __restrict__ w2,    // [H, D] (this expert)
    const float* __restrict__ gate,  // [T, E]
    int e,
    float* __restrict__ out)         // [T, D]
{
    __shared__ float As[2][64][20];   // h tile  [M=64][K=16]
    __shared__ float Bs[2][16][72];   // w2 tile [K=16][N=64]

    const int n0 = blockIdx.x * 64;          // D offset
    const int m0 = blockIdx.y * 64;          // T offset
    const int tid = threadIdx.x;
    const int wave = tid >> 5, lane = tid & 31;
    const int wm = wave & 3;
    const int wn = wave >> 2;
    const int r = lane & 15, half = lane >> 4;

    v8f acc[2];
#pragma unroll
    for (int ns = 0; ns < 2; ++ns)
#pragma unroll
        for (int i = 0; i < 8; ++i) acc[ns][i] = 0.0f;

    const int a_row = tid >> 2, a_col = (tid & 3) * 4;
    const int b_row = tid >> 4, b_col = (tid & 15) * 4;
    const float* gA = h  + (size_t)(m0 + a_row) * H_DIM + a_col;   // +k0
    const float* gB = w2 + (size_t)b_row * D_DIM + n0 + b_col;     // +k0*D
    const unsigned lA[2] = { lds_addr(&As[0][a_row][a_col]),
                             lds_addr(&As[1][a_row][a_col]) };
    const unsigned lB[2] = { lds_addr(&Bs[0][b_row][b_col]),
                             lds_addr(&Bs[1][b_row][b_col]) };

    auto issue = [&](int buf, int k0) {
        async_copy16(lA[buf], gA + k0);
        async_copy16(lB[buf], gB + (size_t)k0 * D_DIM);
    };
    auto compute = [&](int buf) {
#pragma unroll
        for (int kk = 0; kk < 16; kk += 4) {
            v2f a;
            a.x = As[buf][wm * 16 + r][kk + 2 * half];
            a.y = As[buf][wm * 16 + r][kk + 2 * half + 1];
#pragma unroll
            for (int ns = 0; ns < 2; ++ns) {
                const int c = wn * 32 + ns * 16 + r;
                v2f b;
                b.x = Bs[buf][kk + 2 * half][c];
                b.y = Bs[buf][kk + 2 * half + 1][c];
                acc[ns] = wmma_f32_4(a, b, acc[ns]);
            }
        }
    };

    issue(0, 0);
    int k0 = 0;
    for (; k0 + 32 < H_DIM; k0 += 32) {
        issue(1, k0 + 16); WAIT_ASYNC(2);
        __syncthreads(); compute(0); __syncthreads();
        issue(0, k0 + 32); WAIT_ASYNC(2);
        __syncthreads(); compute(1); __syncthreads();
    }
    issue(1, k0 + 16); WAIT_ASYNC(2);
    __syncthreads(); compute(0); __syncthreads();
    WAIT_ASYNC(0);
    __syncthreads(); compute(1);

    // epilogue: scale rows by gate[t, e], accumulate into out
#pragma unroll
    for (int i = 0; i < 8; ++i) {
        const int row = m0 + wm * 16 + i + 8 * half;
        const float g = gate[row * N_EXP + e];
#pragma unroll
        for (int ns = 0; ns < 2; ++ns) {
            const int col = n0 + wn * 32 + ns * 16 + r;
            float* p = out + (size_t)row * D_DIM + col;
            *p += g * acc[ns][i];
        }
    }
}

// ---------------------------------------------------------------------------
// Host-side launch. Workspace layout (needs ~92 MB + 4 MB):
//   [0, T*E*4)            : dense gate
//   [4 MB, 4 MB + T*H*4)  : per-expert h staging (reused across experts)
// ---------------------------------------------------------------------------
extern "C" void kernel_launch(void* const* d_in, const int* in_sizes, int n_in,
                              void* d_out, int out_size, void* d_ws, size_t ws_size,
                              hipStream_t stream) {
    const float* x      = (const float*)d_in[0];
    const float* Wg     = (const float*)d_in[1];
    const float* w1     = (const float*)d_in[2];
    const float* w2     = (const float*)d_in[3];
    const float* w3     = (const float*)d_in[4];
    const float* next_r = (const float*)d_in[5];
    float* out = (float*)d_out;

    float* gate = (float*)d_ws;
    float* h    = (float*)((char*)d_ws + (size_t)(4u << 20));

    // 1) gating
    moe_gate_kernel<<<T_TOK / 8, 256, 0, stream>>>(x, Wg, gate);

    // 2) residual init
    const int n4 = T_TOK * D_DIM / 4;
    moe_init_out<<<(n4 + 255) / 256, 256, 0, stream>>>(
        (const float4*)next_r, (float4*)out, n4);

    // 3) per-expert fused SwiGLU + down-proj accumulate.
    //    Back-to-back launches per expert keep that expert's weights and its
    //    h staging resident in the 192 MB L2.
    for (int e = 0; e < N_EXP; ++e) {
        const float* w1e = w1 + (size_t)e * D_DIM * H_DIM;
        const float* w3e = w3 + (size_t)e * D_DIM * H_DIM;
        const float* w2e = w2 + (size_t)e * H_DIM * D_DIM;

        dim3 gA(H_DIM / 64, T_TOK / 64);
        moe_h_kernel<<<gA, 256, 0, stream>>>(x, w1e, w3e, h);

        dim3 gB(D_DIM / 64, T_TOK / 64);
        moe_out_kernel<<<gB, 256, 0, stream>>>(h, w2e, gate, e, out);
    }
}